// AdditiveAttention_352187319208
// MI455X (gfx1250) — compile-verified
//
#include <hip/hip_runtime.h>
#include <cmath>

#define B_ 16
#define S_ 8192
#define H_ 512
#define KC 8          // K-chunk (floats)
#define PAD 10        // padded LDS row stride: even (8B b64 align), conflict-free
#define NCHUNK (H_ / KC)
#define ASYNC_OPS 5   // async instructions per wave per chunk (4 x b64 W + 1 x b32 enc)

typedef __attribute__((ext_vector_type(2))) float v2f;
typedef __attribute__((ext_vector_type(8))) float v8f;
typedef __attribute__((ext_vector_type(2))) int   v2i;

// Typed address-space pointers for the async global->LDS builtins
typedef __attribute__((address_space(1))) v2i* gptr64_t;
typedef __attribute__((address_space(3))) v2i* lptr64_t;
typedef __attribute__((address_space(1))) int* gptr32_t;
typedef __attribute__((address_space(3))) int* lptr32_t;

#if __has_builtin(__builtin_amdgcn_global_load_async_to_lds_b64) && \
    __has_builtin(__builtin_amdgcn_global_load_async_to_lds_b32)
#define ATT_HAS_ASYNC 1
#else
#define ATT_HAS_ASYNC 0
#endif

#if __has_builtin(__builtin_amdgcn_s_wait_asynccnt)
#define WAIT_ASYNC(n) __builtin_amdgcn_s_wait_asynccnt(n)
#else
#define WAIT_ASYNC(n) asm volatile("s_wait_asynccnt %0" :: "n"(n) : "memory")
#endif

// ---------------- Kernel 1: q_proj[b][o] = dot(query[b,:], W_q[o,:]) ----------------
__global__ __launch_bounds__(256)
void qproj_kernel(const float* __restrict__ query,
                  const float* __restrict__ Wq,
                  float* __restrict__ qp) {
    const int b = blockIdx.x;
    __shared__ float sq[H_];
    for (int i = threadIdx.x; i < H_; i += 256) sq[i] = query[b * H_ + i];
    __syncthreads();
    for (int o = threadIdx.x; o < H_; o += 256) {
        const float* wr = Wq + (size_t)o * H_;
        float s = 0.f;
#pragma unroll 8
        for (int h = 0; h < H_; ++h) s += sq[h] * wr[h];
        qp[b * H_ + o] = s;
    }
}

// ---------------- Kernel 2: fused GEMM + tanh + dot(v) -> raw scores ----------------
// Block: 512 threads = 16 waves. Block owns 64 rows (flat b*S+s).
// Wave (mw, nw) with mw = wave&3, nw = wave>>2: rows [mw*16, mw*16+16), cols [nw*128, nw*128+128).
__global__ __launch_bounds__(512)
void scores_kernel(const float* __restrict__ enc,
                   const float* __restrict__ Wh,
                   const float* __restrict__ qp,
                   const float* __restrict__ v,
                   float* __restrict__ scores) {
    __shared__ float sW[2][H_ * PAD];    // W_h[n][k0..k0+7], padded, double-buffered
    __shared__ float sEnc[2][64 * PAD];  // enc tile 64 x 8, double-buffered
    __shared__ float sQp[H_];
    __shared__ float sV[H_];
    __shared__ float sRed[4 * 4 * 2 * 8]; // [nw][mw][hi][r]

    const int tid  = threadIdx.x;
    const int lane = tid & 31;
    const int wave = tid >> 5;        // 0..15
    const int lo   = lane & 15;
    const int hi   = lane >> 4;
    const int mw   = wave & 3;
    const int nw   = wave >> 2;
    const int rowBase = blockIdx.x * 64;   // flat (b*S + s)
    const int b = rowBase / S_;            // 64 | S -> single b per block

    for (int i = tid; i < H_; i += 512) { sQp[i] = qp[b * H_ + i]; sV[i] = v[i]; }

    v8f acc[8];
#pragma unroll
    for (int t = 0; t < 8; ++t) {
        v8f z = {0.f, 0.f, 0.f, 0.f, 0.f, 0.f, 0.f, 0.f};
        acc[t] = z;
    }

#if ATT_HAS_ASYNC
    // -------- async double-buffered pipeline (GLOBAL_LOAD_ASYNC_TO_LDS_*) --------
    auto stage_async = [&](int buf, int k0) {
#pragma unroll
        for (int i = 0; i < 4; ++i) {
            const int idx = tid + i * 512;     // 0..2047 : 4 b64 per 8-float row
            const int n   = idx >> 2;
            const int kk  = (idx & 3) * 2;
            __builtin_amdgcn_global_load_async_to_lds_b64(
                (gptr64_t)(Wh + (size_t)n * H_ + k0 + kk),
                (lptr64_t)(&sW[buf][n * PAD + kk]), 0, 0);
        }
        const int m = tid >> 3;
        const int k = tid & 7;
        __builtin_amdgcn_global_load_async_to_lds_b32(
            (gptr32_t)(enc + (size_t)(rowBase + m) * H_ + k0 + k),
            (lptr32_t)(&sEnc[buf][m * PAD + k]), 0, 0);
    };

    stage_async(0, 0);
    for (int c = 0; c < NCHUNK; ++c) {
        const int cur = c & 1;
        if (c + 1 < NCHUNK) {
            stage_async(cur ^ 1, (c + 1) * KC);
            WAIT_ASYNC(ASYNC_OPS);   // chunk c complete; chunk c+1 may be in flight
        } else {
            WAIT_ASYNC(0);
        }
        __syncthreads();
#pragma unroll
        for (int kk = 0; kk < KC; kk += 4) {
            v2f a = *reinterpret_cast<const v2f*>(&sEnc[cur][(mw * 16 + lo) * PAD + kk + hi * 2]);
#pragma unroll
            for (int t = 0; t < 8; ++t) {
                const int n = nw * 128 + t * 16 + lo;
                v2f bb = *reinterpret_cast<const v2f*>(&sW[cur][n * PAD + kk + hi * 2]);
                acc[t] = __builtin_amdgcn_wmma_f32_16x16x4_f32(
                    false, a, false, bb, (short)0, acc[t], false, false);
            }
        }
        __syncthreads();   // all waves done reading buffer `cur` before it is refilled
    }
#else
    // -------- fallback: synchronous staging through VGPRs --------
    for (int c = 0; c < NCHUNK; ++c) {
        const int k0 = c * KC;
#pragma unroll
        for (int i = 0; i < 4; ++i) {
            const int idx = tid + i * 512;
            const int n   = idx >> 2;
            const int kk  = (idx & 3) * 2;
            *reinterpret_cast<v2f*>(&sW[0][n * PAD + kk]) =
                *reinterpret_cast<const v2f*>(&Wh[(size_t)n * H_ + k0 + kk]);
        }
        {
            const int m = tid >> 3;
            const int k = tid & 7;
            sEnc[0][m * PAD + k] = enc[(size_t)(rowBase + m) * H_ + k0 + k];
        }
        if (k0 + KC < H_) __builtin_prefetch(&Wh[(size_t)tid * H_ + k0 + KC], 0, 1);
        __syncthreads();
#pragma unroll
        for (int kk = 0; kk < KC; kk += 4) {
            v2f a = *reinterpret_cast<const v2f*>(&sEnc[0][(mw * 16 + lo) * PAD + kk + hi * 2]);
#pragma unroll
            for (int t = 0; t < 8; ++t) {
                const int n = nw * 128 + t * 16 + lo;
                v2f bb = *reinterpret_cast<const v2f*>(&sW[0][n * PAD + kk + hi * 2]);
                acc[t] = __builtin_amdgcn_wmma_f32_16x16x4_f32(
                    false, a, false, bb, (short)0, acc[t], false, false);
            }
        }
        __syncthreads();
    }
#endif

    // Epilogue: p[m] += tanh(h_proj + q_proj[n]) * v[n];  D VGPR r -> M = hi*8+r, N = lo
    float p[8];
#pragma unroll
    for (int r = 0; r < 8; ++r) p[r] = 0.f;
#pragma unroll
    for (int t = 0; t < 8; ++t) {
        const int n = nw * 128 + t * 16 + lo;
        const float qn = sQp[n], vn = sV[n];
#pragma unroll
        for (int r = 0; r < 8; ++r) p[r] += tanhf(acc[t][r] + qn) * vn;
    }
    // Reduce across the 16 `lo` lanes (keep hi) with wave32 shuffles: deterministic.
#pragma unroll
    for (int r = 0; r < 8; ++r) {
#pragma unroll
        for (int msk = 1; msk < 16; msk <<= 1) p[r] += __shfl_xor(p[r], msk, 32);
    }
    if (lo == 0) {
#pragma unroll
        for (int r = 0; r < 8; ++r) sRed[((nw * 4 + mw) * 2 + hi) * 8 + r] = p[r];
    }
    __syncthreads();
    if (tid < 64) {                       // m = tid
        const int mwm = tid >> 4, him = (tid >> 3) & 1, r = tid & 7;
        float s = 0.f;
        for (int nwi = 0; nwi < 4; ++nwi)
            s += sRed[((nwi * 4 + mwm) * 2 + him) * 8 + r];
        scores[rowBase + tid] = s;
    }
}

// ---------------- Kernel 3: softmax over S, in place ----------------
__global__ __launch_bounds__(256)
void softmax_kernel(float* __restrict__ alpha) {
    const int b = blockIdx.x;
    float* row = alpha + (size_t)b * S_;
    __shared__ float red[256];
    const int tid = threadIdx.x;

    float m = -INFINITY;
    for (int i = tid; i < S_; i += 256) m = fmaxf(m, row[i]);
    red[tid] = m; __syncthreads();
    for (int off = 128; off > 0; off >>= 1) {
        if (tid < off) red[tid] = fmaxf(red[tid], red[tid + off]);
        __syncthreads();
    }
    const float mx = red[0]; __syncthreads();

    float s = 0.f;
    for (int i = tid; i < S_; i += 256) s += __expf(row[i] - mx);
    red[tid] = s; __syncthreads();
    for (int off = 128; off > 0; off >>= 1) {
        if (tid < off) red[tid] += red[tid + off];
        __syncthreads();
    }
    const float inv = 1.f / red[0];
    for (int i = tid; i < S_; i += 256) row[i] = __expf(row[i] - mx) * inv;
}

// ---------------- Kernel 4: partial context over S-chunks (deterministic) ----------------
__global__ __launch_bounds__(512)
void ctx_partial_kernel(const float* __restrict__ enc,
                        const float* __restrict__ alpha,
                        float* __restrict__ part) {
    const int blk = blockIdx.x;
    const int b  = blk >> 5;
    const int sc = blk & 31;
    const int h  = threadIdx.x;
    __shared__ float sA[256];
    const int s0 = sc * 256;
    if (threadIdx.x < 256) sA[threadIdx.x] = alpha[(size_t)b * S_ + s0 + threadIdx.x];
    __syncthreads();
    float acc = 0.f;
    const float* e = enc + ((size_t)b * S_ + s0) * H_ + h;
    for (int s = 0; s < 256; ++s) acc += sA[s] * e[(size_t)s * H_];
    part[(size_t)blk * H_ + h] = acc;
}

// ---------------- Kernel 5: reduce partials -> context ----------------
__global__ __launch_bounds__(512)
void ctx_reduce_kernel(const float* __restrict__ part, float* __restrict__ ctx) {
    const int b = blockIdx.x;
    const int h = threadIdx.x;
    float s = 0.f;
    for (int sc = 0; sc < 32; ++sc) s += part[(size_t)(b * 32 + sc) * H_ + h];
    ctx[b * H_ + h] = s;
}

extern "C" void kernel_launch(void* const* d_in, const int* in_sizes, int n_in,
                              void* d_out, int out_size, void* d_ws, size_t ws_size,
                              hipStream_t stream) {
    const float* enc   = (const float*)d_in[0];  // (B,S,H)
    const float* query = (const float*)d_in[1];  // (B,H)
    const float* Wh    = (const float*)d_in[2];  // (H,H)
    const float* Wq    = (const float*)d_in[3];  // (H,H)
    const float* v     = (const float*)d_in[4];  // (H)

    float* ctx   = (float*)d_out;                // (B,H)
    float* alpha = (float*)d_out + B_ * H_;      // (B,S)  scores -> softmax in place

    float* qp   = (float*)d_ws;                  // B*H floats
    float* part = (float*)d_ws + B_ * H_;        // (B*32)*H floats

    qproj_kernel<<<B_, 256, 0, stream>>>(query, Wq, qp);
    scores_kernel<<<(B_ * S_) / 64, 512, 0, stream>>>(enc, Wh, qp, v, alpha);
    softmax_kernel<<<B_, 256, 0, stream>>>(alpha);
    ctx_partial_kernel<<<B_ * 32, 512, 0, stream>>>(enc, alpha, part);
    ctx_reduce_kernel<<<B_, 512, 0, stream>>>(part, ctx);
}